// SharedGNN_88940182766121
// MI455X (gfx1250) — compile-verified
//
#include <hip/hip_runtime.h>
#include <hip/hip_bf16.h>

typedef __attribute__((ext_vector_type(2))) float v2f;
typedef __attribute__((ext_vector_type(8))) float v8f;

#define FD 64        // feature dim D
#define NG 64        // graphs G

// ---------------- degree accumulation ----------------
__global__ __launch_bounds__(256) void deg_kernel(const int* __restrict__ src,
                                                  const int* __restrict__ dst,
                                                  float* __restrict__ outdeg,
                                                  float* __restrict__ indeg, int E) {
    int e = blockIdx.x * blockDim.x + threadIdx.x;
    if (e < E) {
        atomicAdd(&outdeg[src[e]], 1.0f);
        atomicAdd(&indeg[dst[e]], 1.0f);
    }
}

// deg -> max(deg,1)^-0.5 (applied to both arrays, contiguous 2N)
__global__ __launch_bounds__(256) void norm_kernel(float* __restrict__ buf, int n2) {
    int i = blockIdx.x * blockDim.x + threadIdx.x;
    if (i < n2) buf[i] = rsqrtf(fmaxf(buf[i], 1.0f));
}

// ---------------- embedding gather: h[n] = emb[node_feat[n]] ----------------
__global__ __launch_bounds__(256) void embed_kernel(const int* __restrict__ node_feat,
                                                    const float* __restrict__ emb,
                                                    float* __restrict__ h, int N) {
    int tid = blockIdx.x * blockDim.x + threadIdx.x;
    int n = tid >> 5;
    int c = (tid & 31) * 2;
    if (n < N) {
        int t = node_feat[n];
        float2 v = *reinterpret_cast<const float2*>(emb + (size_t)t * FD + c);
        *reinterpret_cast<float2*>(h + (size_t)n * FD + c) = v;
    }
}

// ---------------- edge scatter: agg[dst] += h[src] * out_norm[src] ----------------
// one wave32 per edge; each lane handles 2 adjacent columns (float2 read, 2 f32 atomics)
__global__ __launch_bounds__(256) void scatter_kernel(const int* __restrict__ src,
                                                      const int* __restrict__ dst,
                                                      const float* __restrict__ h,
                                                      const float* __restrict__ out_norm,
                                                      float* __restrict__ agg, int E) {
    int tid = blockIdx.x * blockDim.x + threadIdx.x;
    int e = tid >> 5;
    int c = (tid & 31) * 2;
    if (e < E) {
        int s = src[e];
        int d = dst[e];
        float sc = out_norm[s];
        float2 v = *reinterpret_cast<const float2*>(h + (size_t)s * FD + c);
        atomicAdd(&agg[(size_t)d * FD + c],     v.x * sc);
        atomicAdd(&agg[(size_t)d * FD + c + 1], v.y * sc);
    }
}

// ---------------- fused scale + GEMM + bias + ReLU via V_WMMA_F32_16X16X4_F32 ----
// Y[m][:] = relu( (agg[m][:] * in_norm[m]) @ W + b ),  W is 64x64 row-major [k][n].
// One wave computes a 16-row strip x 64 cols: 4 accumulator tiles, K swept in steps of 4.
// W is staged in LDS pre-swizzled into K-pairs: sWp[kpair][col][2] = {W[2kp][col], W[2kp+1][col]}
// so every B fragment is one aligned ds_load_b64 straight into an even VGPR pair.
__global__ __launch_bounds__(256) void gemm_wmma_kernel(const float* __restrict__ agg,
                                                        const float* __restrict__ in_norm,
                                                        const float* __restrict__ W,
                                                        const float* __restrict__ bias,
                                                        float* __restrict__ hout, int N) {
    __shared__ float sWp[FD * FD];   // [32 kpairs][64 cols][2]
    __shared__ float sB[FD];
    int t = threadIdx.x;             // 0..255, 8 waves
    for (int i = t; i < FD * FD; i += 256) {
        int k   = i >> 6;
        int col = i & 63;
        sWp[(k >> 1) * 128 + col * 2 + (k & 1)] = W[i];
    }
    if (t < FD) sB[t] = bias[t];
    __syncthreads();

    int wave = t >> 5;
    int lane = t & 31;
    int strip = blockIdx.x * 8 + wave;   // 16-row strip index
    int row0 = strip * 16;
    if (row0 >= N) return;               // strip-uniform: EXEC stays all-ones for WMMA

    int lh = lane & 15;                  // M (and N) sub-index
    int lg = lane >> 4;                  // half-wave select
    int m  = row0 + lh;
    if (m >= N) m = N - 1;               // clamp (N is a multiple of 16 here, no-op)
    float s = in_norm[m];
    const float* arow = agg + (size_t)m * FD;

    v8f acc0 = {}, acc1 = {}, acc2 = {}, acc3 = {};
#pragma unroll
    for (int k4 = 0; k4 < FD / 4; ++k4) {
        int kb = k4 * 4 + 2 * lg;        // A layout: v0 lanes0-15 K=0 / lanes16-31 K=2; v1: K=1/K=3
        v2f a;
        a.x = arow[kb]     * s;
        a.y = arow[kb + 1] * s;
        // B fragments: kpair = kb/2 = k4*2 + lg; contiguous {W[kb][col], W[kb+1][col]}
        const float* wp = &sWp[(k4 * 2 + lg) * 128 + lh * 2];
        v2f b0 = *reinterpret_cast<const v2f*>(wp);
        v2f b1 = *reinterpret_cast<const v2f*>(wp + 32);
        v2f b2 = *reinterpret_cast<const v2f*>(wp + 64);
        v2f b3 = *reinterpret_cast<const v2f*>(wp + 96);
        acc0 = __builtin_amdgcn_wmma_f32_16x16x4_f32(false, a, false, b0, (short)0, acc0, false, false);
        acc1 = __builtin_amdgcn_wmma_f32_16x16x4_f32(false, a, false, b1, (short)0, acc1, false, false);
        acc2 = __builtin_amdgcn_wmma_f32_16x16x4_f32(false, a, false, b2, (short)0, acc2, false, false);
        acc3 = __builtin_amdgcn_wmma_f32_16x16x4_f32(false, a, false, b3, (short)0, acc3, false, false);
    }

    // D layout: VGPR r -> M = row0 + r + 8*lg, N = lh (+16/32/48 per tile)
#pragma unroll
    for (int r = 0; r < 8; ++r) {
        int mr = row0 + r + 8 * lg;
        if (mr < N) {
            float* orow = hout + (size_t)mr * FD;
            orow[lh]      = fmaxf(acc0[r] + sB[lh],      0.0f);
            orow[16 + lh] = fmaxf(acc1[r] + sB[16 + lh], 0.0f);
            orow[32 + lh] = fmaxf(acc2[r] + sB[32 + lh], 0.0f);
            orow[48 + lh] = fmaxf(acc3[r] + sB[48 + lh], 0.0f);
        }
    }
}

// ---------------- per-graph mean pooling ----------------
__global__ __launch_bounds__(256) void pool_kernel(const int* __restrict__ graph_ids,
                                                   const float* __restrict__ h,
                                                   float* __restrict__ hg,
                                                   float* __restrict__ counts, int N) {
    int tid = blockIdx.x * blockDim.x + threadIdx.x;
    int n = tid >> 5;
    int lane = tid & 31;
    int c = lane * 2;
    if (n < N) {
        int g = graph_ids[n];
        float2 v = *reinterpret_cast<const float2*>(h + (size_t)n * FD + c);
        atomicAdd(&hg[g * FD + c],     v.x);
        atomicAdd(&hg[g * FD + c + 1], v.y);
        if (lane == 0) atomicAdd(&counts[g], 1.0f);
    }
}

__global__ __launch_bounds__(256) void div_kernel(float* __restrict__ hg,
                                                  const float* __restrict__ counts, int GD) {
    int i = blockIdx.x * blockDim.x + threadIdx.x;
    if (i < GD) hg[i] = hg[i] / fmaxf(counts[i >> 6], 1.0f);
}

extern "C" void kernel_launch(void* const* d_in, const int* in_sizes, int n_in,
                              void* d_out, int out_size, void* d_ws, size_t ws_size,
                              hipStream_t stream) {
    (void)n_in; (void)out_size; (void)ws_size;
    const int*   node_feat = (const int*)d_in[0];
    const int*   edge_idx  = (const int*)d_in[1];
    const int*   graph_ids = (const int*)d_in[2];
    const float* emb       = (const float*)d_in[3];
    const float* Ws        = (const float*)d_in[4];
    const float* bs        = (const float*)d_in[5];
    float*       out       = (float*)d_out;

    const int N = in_sizes[0];
    const int E = in_sizes[1] / 2;
    const int L = in_sizes[4] / (FD * FD);

    const int* src = edge_idx;
    const int* dst = edge_idx + E;

    // workspace layout (floats): out_norm[N] | in_norm[N] | h[N*FD] | agg[N*FD] | counts[NG]
    float* ws       = (float*)d_ws;
    float* out_norm = ws;
    float* in_norm  = ws + N;
    float* h        = ws + 2 * (size_t)N;
    float* agg      = h + (size_t)N * FD;
    float* counts   = agg + (size_t)N * FD;

    hipMemsetAsync(out_norm, 0, sizeof(float) * 2 * (size_t)N, stream);
    hipMemsetAsync(counts,   0, sizeof(float) * NG, stream);
    hipMemsetAsync(out,      0, sizeof(float) * NG * FD, stream);

    deg_kernel <<<(E + 255) / 256, 256, 0, stream>>>(src, dst, out_norm, in_norm, E);
    norm_kernel<<<(2 * N + 255) / 256, 256, 0, stream>>>(out_norm, 2 * N);
    embed_kernel<<<((size_t)N * 32 + 255) / 256, 256, 0, stream>>>(node_feat, emb, h, N);

    int strips = (N + 15) / 16;
    int gemm_blocks = (strips + 7) / 8;
    int scatter_blocks = (int)(((size_t)E * 32 + 255) / 256);

    for (int l = 0; l < L; ++l) {
        hipMemsetAsync(agg, 0, sizeof(float) * (size_t)N * FD, stream);
        scatter_kernel<<<scatter_blocks, 256, 0, stream>>>(src, dst, h, out_norm, agg, E);
        gemm_wmma_kernel<<<gemm_blocks, 256, 0, stream>>>(agg, in_norm,
                                                          Ws + (size_t)l * FD * FD,
                                                          bs + (size_t)l * FD, h, N);
    }

    pool_kernel<<<((size_t)N * 32 + 255) / 256, 256, 0, stream>>>(graph_ids, h, out, counts, N);
    div_kernel<<<(NG * FD + 255) / 256, 256, 0, stream>>>(out, counts, NG * FD);
}